// SSMrPPG_77446850282143
// MI455X (gfx1250) — compile-verified
//
#include <hip/hip_runtime.h>
#include <math.h>

// ---------------- dims ----------------
#define BATCH   4
#define SEQ     4096
#define DMODEL  256
#define DINNER  512
#define NHEADS  8
#define HEADDIM 64
#define DSTATE  16
#define CHUNK   64
#define NCHUNK  (SEQ / CHUNK)       // 64
#define CONVDIM 544                 // DINNER + 2*DSTATE
#define DPROJ   1576                // 3*DINNER + 2*DSTATE + NHEADS
#define NPAD    1600                // DPROJ padded to multiple of 64
#define TOKENS  (BATCH * SEQ)       // 16384

typedef float v2f __attribute__((ext_vector_type(2)));
typedef float v8f __attribute__((ext_vector_type(8)));

__device__ __forceinline__ v8f wmma_f32(v2f a, v2f b, v8f c) {
  // D = A(16x4) x B(4x16) + C, fp32, wave32
  return __builtin_amdgcn_wmma_f32_16x16x4_f32(false, a, false, b, (short)0, c,
                                               false, false);
}

// Async global->LDS copy (CDNA5 GLOBAL_LOAD_ASYNC_TO_LDS_B32, ASYNCcnt).
// LDS operand = low 32 bits of the generic address (LDS aperture offset).
__device__ __forceinline__ void async_copy_f32(const float* gptr, float* lptr) {
  unsigned lofs = (unsigned)(unsigned long long)lptr;
  unsigned long long ga = (unsigned long long)gptr;
  asm volatile("global_load_async_to_lds_b32 %0, %1, off"
               :: "v"(lofs), "v"(ga)
               : "memory");
}
__device__ __forceinline__ void async_wait0() {
  asm volatile("s_wait_asynccnt 0x0" ::: "memory");
}

// ---------------- K0: pad W_in (256 x 1576 -> 256 x 1600) ----------------
__global__ void k_pad(const float* __restrict__ W, float* __restrict__ Wp) {
  int i = blockIdx.x * 256 + threadIdx.x;
  if (i >= DMODEL * NPAD) return;
  int rw = i / NPAD, cl = i % NPAD;
  Wp[i] = (cl < DPROJ) ? W[rw * DPROJ + cl] : 0.f;
}

// ---------------- K1/K7: WMMA fp32 GEMM  C[M x N] = X[M x K] @ W[K x N] ---
// grid: (N/64, M/64), block 256 (8 waves). Each block: 64x64 tile,
// each wave: 16x32 (two 16x16 accumulators). K staged in 32-wide LDS slabs
// via async global->LDS copies (no VGPR round-trip).
__global__ __launch_bounds__(256) void k_gemm(const float* __restrict__ X,
                                              const float* __restrict__ W,
                                              float* __restrict__ C_,
                                              int K, int N) {
  __shared__ float sA[64 * 33];
  __shared__ float sB[32 * 68];
  const int tid = threadIdx.x;
  const int m0 = blockIdx.y * 64;
  const int n0 = blockIdx.x * 64;
  const int wv = tid >> 5, lane = tid & 31, hl = lane >> 4, r = lane & 15;
  const int tm = (wv & 3) * 16;
  const int tn = (wv >> 2) * 32;
  v8f acc0 = {}, acc1 = {};
  for (int k0 = 0; k0 < K; k0 += 32) {
    __syncthreads();
    for (int i = tid; i < 64 * 32; i += 256) {
      int rr = i >> 5, cc = i & 31;
      async_copy_f32(&X[(size_t)(m0 + rr) * K + k0 + cc], &sA[rr * 33 + cc]);
    }
    for (int i = tid; i < 32 * 64; i += 256) {
      int rr = i >> 6, cc = i & 63;
      async_copy_f32(&W[(size_t)(k0 + rr) * N + n0 + cc], &sB[rr * 68 + cc]);
    }
    async_wait0();
    __syncthreads();
#pragma unroll
    for (int kk = 0; kk < 32; kk += 4) {
      v2f a, b0, b1;
      a.x  = sA[(tm + r) * 33 + kk + 2 * hl];
      a.y  = sA[(tm + r) * 33 + kk + 2 * hl + 1];
      b0.x = sB[(kk + 2 * hl) * 68 + tn + r];
      b0.y = sB[(kk + 2 * hl + 1) * 68 + tn + r];
      b1.x = sB[(kk + 2 * hl) * 68 + tn + 16 + r];
      b1.y = sB[(kk + 2 * hl + 1) * 68 + tn + 16 + r];
      acc0 = wmma_f32(a, b0, acc0);
      acc1 = wmma_f32(a, b1, acc1);
    }
  }
#pragma unroll
  for (int v = 0; v < 8; ++v) {
    int row = m0 + tm + v + 8 * hl;
    C_[(size_t)row * N + n0 + tn + r]      = acc0[v];
    C_[(size_t)row * N + n0 + tn + 16 + r] = acc1[v];
  }
}

// ---------------- K2: causal depthwise conv (K=4) + ReLU6 + dt ------------
__global__ void k_conv(const float* __restrict__ zxb,
                       const float* __restrict__ conv_w,
                       const float* __restrict__ conv_b,
                       const float* __restrict__ initc,
                       const float* __restrict__ dt_bias,
                       float* __restrict__ xin, float* __restrict__ Bm,
                       float* __restrict__ Cm, float* __restrict__ dtb) {
  int idx = blockIdx.x * 256 + threadIdx.x;
  const int TOT = TOKENS * (CONVDIM + NHEADS);
  if (idx >= TOT) return;
  int tok = idx / (CONVDIM + NHEADS), c = idx % (CONVDIM + NHEADS);
  int b = tok >> 12, t = tok & 4095;
  if (c < CONVDIM) {
    float acc = conv_b[c];
#pragma unroll
    for (int j = 0; j < 4; ++j) {
      int s = t + j - 3;
      float u = (s >= 0)
                    ? zxb[(size_t)((b << 12) + s) * NPAD + 2 * DINNER + c]
                    : initc[(s + 4) * CONVDIM + c];
      acc += conv_w[j * CONVDIM + c] * u;
    }
    acc = fminf(fmaxf(acc, 0.f), 6.f);
    if (c < DINNER)           xin[(size_t)tok * DINNER + c] = acc;
    else if (c < DINNER + 16) Bm[tok * 16 + (c - DINNER)] = acc;
    else                      Cm[tok * 16 + (c - DINNER - 16)] = acc;
  } else {
    int h = c - CONVDIM;
    float d =
        fminf(expf(zxb[(size_t)tok * NPAD + 3 * DINNER + 2 * DSTATE + h] +
                   dt_bias[h]),
              6.f);
    dtb[tok * 8 + h] = d;
  }
}

// ---------------- K3: per (b, chunk, head) intra-chunk SSD ----------------
__global__ __launch_bounds__(256) void k_chunk(
    const float* __restrict__ xin, const float* __restrict__ Bm,
    const float* __restrict__ Cm, const float* __restrict__ dtb,
    const float* __restrict__ A_log, float* __restrict__ acum_re,
    float* __restrict__ acum_im, float* __restrict__ cs_re,
    float* __restrict__ cs_im, float* __restrict__ S_re,
    float* __restrict__ S_im, float* __restrict__ y2) {
  __shared__ float sX[64 * 65];
  __shared__ float sCB[64 * 65];
  __shared__ float sBt[64 * 17];
  __shared__ float sCt[64 * 17];
  __shared__ float ac_re[64], ac_im[64], dre[64], dimv[64], tre[64], tim[64];
  const int tid = threadIdx.x;
  const int c = blockIdx.x, h = blockIdx.y, b = blockIdx.z;
  const int row0 = (b << 12) + (c << 6);
  const int wv = tid >> 5, lane = tid & 31, hl = lane >> 4, r = lane & 15;

  // async-stage B and C tiles into LDS
  for (int i = tid; i < 64 * 16; i += 256) {
    int l = i >> 4, n = i & 15;
    async_copy_f32(&Bm[(size_t)(row0 + l) * 16 + n], &sBt[l * 17 + n]);
    async_copy_f32(&Cm[(size_t)(row0 + l) * 16 + n], &sCt[l * 17 + n]);
  }
  float A_re = -expf(A_log[h * 2 + 0]);
  float A_im = A_log[h * 2 + 1];
  if (tid < 64) {
    float dtl = dtb[(row0 + tid) * 8 + h];
    tre[tid] = A_re * dtl;
    tim[tid] = A_im * dtl;
  }
  for (int i = tid; i < 64 * 64; i += 256) {
    int l = i >> 6, p = i & 63;
    sX[l * 65 + p] =
        xin[(size_t)(row0 + l) * DINNER + h * 64 + p] * dtb[(row0 + l) * 8 + h];
  }
  async_wait0();
  __syncthreads();
  if (tid == 0) {
    float cr = 0.f, ci = 0.f;
    for (int l = 0; l < 64; ++l) {
      cr += tre[l]; ci += tim[l];
      ac_re[l] = cr; ac_im[l] = ci;
    }
  }
  __syncthreads();
  if (tid < 64) {
    acum_re[(row0 + tid) * 8 + h] = ac_re[tid];
    acum_im[(row0 + tid) * 8 + h] = ac_im[tid];
    float sc = expf(ac_re[63] - ac_re[tid]);
    float an = ac_im[63] - ac_im[tid];
    dre[tid] = sc * cosf(an);
    dimv[tid] = sc * sinf(an);
  }
  if (tid == 0) {
    int bch = ((b << 6) + c) * 8 + h;
    S_re[bch] = ac_re[63];
    S_im[bch] = ac_im[63];
  }
  __syncthreads();
  // GEMM1: CB(64x64) = C(64x16) @ B^T(16x64)
  {
    const int tm = (wv & 3) * 16;
    const int tn = (wv >> 2) * 32;
    v8f a0 = {}, a1 = {};
#pragma unroll
    for (int kk = 0; kk < 16; kk += 4) {
      v2f a, b0, b1;
      a.x  = sCt[(tm + r) * 17 + kk + 2 * hl];
      a.y  = sCt[(tm + r) * 17 + kk + 2 * hl + 1];
      b0.x = sBt[(tn + r) * 17 + kk + 2 * hl];
      b0.y = sBt[(tn + r) * 17 + kk + 2 * hl + 1];
      b1.x = sBt[(tn + 16 + r) * 17 + kk + 2 * hl];
      b1.y = sBt[(tn + 16 + r) * 17 + kk + 2 * hl + 1];
      a0 = wmma_f32(a, b0, a0);
      a1 = wmma_f32(a, b1, a1);
    }
#pragma unroll
    for (int v = 0; v < 8; ++v) {
      int rr = tm + v + 8 * hl;
      sCB[rr * 65 + tn + r]      = a0[v];
      sCB[rr * 65 + tn + 16 + r] = a1[v];
    }
  }
  __syncthreads();
  // GEMM2: Y_re/Y_im = (CB .* Lmat) @ xc ; decay mask built into A fragments
  {
    const int tm = (wv & 3) * 16;
    const int tn = (wv >> 2) * 32;
    v8f yr0 = {}, yr1 = {}, yi0 = {}, yi1 = {};
    const int lrow = tm + r;
    const float alr = ac_re[lrow], ali = ac_im[lrow];
#pragma unroll 4
    for (int kk = 0; kk < 64; kk += 4) {
      int s0 = kk + 2 * hl, s1 = s0 + 1;
      v2f ar, ai, b0, b1;
      {
        float raw = sCB[lrow * 65 + s0];
        float er = expf(alr - ac_re[s0]);
        float an = ali - ac_im[s0];
        bool ok = lrow >= s0;
        ar.x = ok ? raw * er * cosf(an) : 0.f;
        ai.x = ok ? raw * er * sinf(an) : 0.f;
      }
      {
        float raw = sCB[lrow * 65 + s1];
        float er = expf(alr - ac_re[s1]);
        float an = ali - ac_im[s1];
        bool ok = lrow >= s1;
        ar.y = ok ? raw * er * cosf(an) : 0.f;
        ai.y = ok ? raw * er * sinf(an) : 0.f;
      }
      b0.x = sX[s0 * 65 + tn + r];
      b0.y = sX[s1 * 65 + tn + r];
      b1.x = sX[s0 * 65 + tn + 16 + r];
      b1.y = sX[s1 * 65 + tn + 16 + r];
      yr0 = wmma_f32(ar, b0, yr0);
      yr1 = wmma_f32(ar, b1, yr1);
      yi0 = wmma_f32(ai, b0, yi0);
      yi1 = wmma_f32(ai, b1, yi1);
    }
#pragma unroll
    for (int v = 0; v < 8; ++v) {
      int l = tm + v + 8 * hl;
      size_t rowoff = (size_t)(row0 + l) * 1024;
      y2[rowoff + (2 * h) * 64 + tn + r]          = yr0[v];
      y2[rowoff + (2 * h) * 64 + tn + 16 + r]     = yr1[v];
      y2[rowoff + (2 * h + 1) * 64 + tn + r]      = yi0[v];
      y2[rowoff + (2 * h + 1) * 64 + tn + 16 + r] = yi1[v];
    }
  }
  // GEMM3: chunk states(64x16) = xc^T @ (B .* decay), re (waves 0-3) / im
  {
    const int tm = (wv & 3) * 16;
    const bool isIm = wv >= 4;
    v8f st = {};
#pragma unroll 4
    for (int kk = 0; kk < 64; kk += 4) {
      int l0 = kk + 2 * hl, l1 = l0 + 1;
      v2f a, bb;
      a.x = sX[l0 * 65 + tm + r];
      a.y = sX[l1 * 65 + tm + r];
      float f0 = isIm ? dimv[l0] : dre[l0];
      float f1 = isIm ? dimv[l1] : dre[l1];
      bb.x = sBt[l0 * 17 + r] * f0;
      bb.y = sBt[l1 * 17 + r] * f1;
      st = wmma_f32(a, bb, st);
    }
    size_t base = (size_t)(((b << 6) + c) * 8 + h) * 1024;
    float* dst = isIm ? cs_im : cs_re;
#pragma unroll
    for (int v = 0; v < 8; ++v) {
      int p = tm + v + 8 * hl;
      dst[base + p * 16 + r] = st[v];
    }
  }
}

// ---------------- K4: sequential inter-chunk complex scan -----------------
__global__ __launch_bounds__(256) void k_scan(
    const float* __restrict__ init_ssm, const float* __restrict__ cs_re,
    const float* __restrict__ cs_im, const float* __restrict__ S_re,
    const float* __restrict__ S_im, float* __restrict__ H_re,
    float* __restrict__ H_im) {
  int h = blockIdx.x, b = blockIdx.y;
  int tid = threadIdx.x;
  float hr[4], hi[4];
#pragma unroll
  for (int k = 0; k < 4; ++k) {
    int e = tid + 256 * k;
    hr[k] = init_ssm[(size_t)(h * 1024 + e) * 2 + 0];
    hi[k] = init_ssm[(size_t)(h * 1024 + e) * 2 + 1];
  }
  for (int c = 0; c < NCHUNK; ++c) {
    int bch = ((b << 6) + c) * 8 + h;
    size_t base = (size_t)bch * 1024;
    float er = expf(S_re[bch]);
    float ecr = er * cosf(S_im[bch]);
    float esi = er * sinf(S_im[bch]);
#pragma unroll
    for (int k = 0; k < 4; ++k) {
      int e = tid + 256 * k;
      H_re[base + e] = hr[k];
      H_im[base + e] = hi[k];
      float nr = ecr * hr[k] - esi * hi[k] + cs_re[base + e];
      float ni = ecr * hi[k] + esi * hr[k] + cs_im[base + e];
      hr[k] = nr; hi[k] = ni;
    }
  }
}

// ---------------- K5: Y_off = exp(A_cum) * (C @ H^T), accumulate into y2 --
__global__ __launch_bounds__(256) void k_yoff(
    const float* __restrict__ Cm, const float* __restrict__ H_re,
    const float* __restrict__ H_im, const float* __restrict__ acum_re,
    const float* __restrict__ acum_im, float* __restrict__ y2) {
  __shared__ float sCt[64 * 17];
  __shared__ float sHr[64 * 17];
  __shared__ float sHi[64 * 17];
  __shared__ float sSr[64 * 65];
  __shared__ float sSi[64 * 65];
  const int tid = threadIdx.x;
  const int c = blockIdx.x, h = blockIdx.y, b = blockIdx.z;
  const int row0 = (b << 12) + (c << 6);
  const size_t hbase = (size_t)(((b << 6) + c) * 8 + h) * 1024;
  const int wv = tid >> 5, lane = tid & 31, hl = lane >> 4, r = lane & 15;
  for (int i = tid; i < 64 * 16; i += 256) {
    int p = i >> 4, n = i & 15;
    async_copy_f32(&Cm[(size_t)(row0 + p) * 16 + n], &sCt[p * 17 + n]);
    async_copy_f32(&H_re[hbase + p * 16 + n], &sHr[p * 17 + n]);
    async_copy_f32(&H_im[hbase + p * 16 + n], &sHi[p * 17 + n]);
  }
  async_wait0();
  __syncthreads();
  {
    const int tm = (wv & 3) * 16;
    const bool isIm = wv >= 4;
    const float* sH = isIm ? sHi : sHr;
    float* sS = isIm ? sSi : sSr;
    v8f acc[4] = {{}, {}, {}, {}};
#pragma unroll
    for (int kk = 0; kk < 16; kk += 4) {
      v2f a;
      a.x = sCt[(tm + r) * 17 + kk + 2 * hl];
      a.y = sCt[(tm + r) * 17 + kk + 2 * hl + 1];
#pragma unroll
      for (int ct = 0; ct < 4; ++ct) {
        v2f bb;
        bb.x = sH[(ct * 16 + r) * 17 + kk + 2 * hl];
        bb.y = sH[(ct * 16 + r) * 17 + kk + 2 * hl + 1];
        acc[ct] = wmma_f32(a, bb, acc[ct]);
      }
    }
#pragma unroll
    for (int ct = 0; ct < 4; ++ct)
#pragma unroll
      for (int v = 0; v < 8; ++v)
        sS[(tm + v + 8 * hl) * 65 + ct * 16 + r] = acc[ct][v];
  }
  __syncthreads();
  for (int i = tid; i < 64 * 64; i += 256) {
    int l = i >> 6, p = i & 63;
    float are = acum_re[(row0 + l) * 8 + h];
    float aim = acum_im[(row0 + l) * 8 + h];
    float er = expf(are);
    float Er = er * cosf(aim), Ei = er * sinf(aim);
    float Sr = sSr[l * 65 + p], Si = sSi[l * 65 + p];
    size_t rowoff = (size_t)(row0 + l) * 1024;
    y2[rowoff + (2 * h) * 64 + p]     += Er * Sr - Ei * Si;
    y2[rowoff + (2 * h + 1) * 64 + p] += Er * Si + Ei * Sr;
  }
}

// ---------------- K6: D skip + gating + RMS norm (in place on y2) ---------
__global__ __launch_bounds__(256) void k_gate(const float* __restrict__ zxb,
                                              const float* __restrict__ xin,
                                              const float* __restrict__ Dv,
                                              const float* __restrict__ norm_w,
                                              float* __restrict__ y2) {
  __shared__ float red[256];
  int t = blockIdx.x;
  int tid = threadIdx.x;
  float yz[4];
  float ss = 0.f;
#pragma unroll
  for (int k = 0; k < 4; ++k) {
    int e = tid + 256 * k;
    int hh = e >> 6, p = e & 63;
    float v = y2[(size_t)t * 1024 + e];
    if ((hh & 1) == 0) v += xin[(size_t)t * DINNER + (hh >> 1) * 64 + p] * Dv[hh];
    float z = zxb[(size_t)t * NPAD + e];
    z = fminf(fmaxf(z, 0.f), 6.f);
    yz[k] = v * z;
    ss += yz[k] * yz[k];
  }
  red[tid] = ss;
  __syncthreads();
  for (int s = 128; s > 0; s >>= 1) {
    if (tid < s) red[tid] += red[tid + s];
    __syncthreads();
  }
  float scale = rsqrtf(red[0] / 1024.f + 1e-5f);
#pragma unroll
  for (int k = 0; k < 4; ++k) {
    int e = tid + 256 * k;
    y2[(size_t)t * 1024 + e] = yz[k] * scale * norm_w[e];
  }
}

// ---------------- launcher ----------------
extern "C" void kernel_launch(void* const* d_in, const int* in_sizes, int n_in,
                              void* d_out, int out_size, void* d_ws,
                              size_t ws_size, hipStream_t stream) {
  const float* x        = (const float*)d_in[0];
  const float* W_in     = (const float*)d_in[1];
  const float* conv_w   = (const float*)d_in[2];
  const float* conv_b   = (const float*)d_in[3];
  const float* A_log    = (const float*)d_in[4];
  const float* dt_bias  = (const float*)d_in[5];
  const float* Dv       = (const float*)d_in[6];
  const float* initc    = (const float*)d_in[7];
  const float* init_ssm = (const float*)d_in[8];
  const float* norm_w   = (const float*)d_in[9];
  const float* W_out    = (const float*)d_in[10];
  float* out = (float*)d_out;

  float* ws = (float*)d_ws;
  size_t o = 0;
  float* w_in_pad = ws + o; o += (size_t)DMODEL * NPAD;
  float* zxb      = ws + o; o += (size_t)TOKENS * NPAD;
  float* xin      = ws + o; o += (size_t)TOKENS * DINNER;
  float* Bm       = ws + o; o += (size_t)TOKENS * 16;
  float* Cm       = ws + o; o += (size_t)TOKENS * 16;
  float* dtb      = ws + o; o += (size_t)TOKENS * 8;
  float* acr      = ws + o; o += (size_t)TOKENS * 8;
  float* aci      = ws + o; o += (size_t)TOKENS * 8;
  float* csr      = ws + o; o += (size_t)BATCH * NCHUNK * NHEADS * 1024;
  float* csi      = ws + o; o += (size_t)BATCH * NCHUNK * NHEADS * 1024;
  float* Sre      = ws + o; o += (size_t)BATCH * NCHUNK * NHEADS;
  float* Sim      = ws + o; o += (size_t)BATCH * NCHUNK * NHEADS;
  float* Hre      = ws + o; o += (size_t)BATCH * NCHUNK * NHEADS * 1024;
  float* Him      = ws + o; o += (size_t)BATCH * NCHUNK * NHEADS * 1024;
  float* y2       = ws + o; o += (size_t)TOKENS * 1024;

  k_pad<<<(DMODEL * NPAD + 255) / 256, 256, 0, stream>>>(W_in, w_in_pad);
  k_gemm<<<dim3(NPAD / 64, TOKENS / 64), 256, 0, stream>>>(x, w_in_pad, zxb,
                                                           DMODEL, NPAD);
  k_conv<<<(TOKENS * (CONVDIM + NHEADS) + 255) / 256, 256, 0, stream>>>(
      zxb, conv_w, conv_b, initc, dt_bias, xin, Bm, Cm, dtb);
  k_chunk<<<dim3(NCHUNK, NHEADS, BATCH), 256, 0, stream>>>(
      xin, Bm, Cm, dtb, A_log, acr, aci, csr, csi, Sre, Sim, y2);
  k_scan<<<dim3(NHEADS, BATCH), 256, 0, stream>>>(init_ssm, csr, csi, Sre, Sim,
                                                  Hre, Him);
  k_yoff<<<dim3(NCHUNK, NHEADS, BATCH), 256, 0, stream>>>(Cm, Hre, Him, acr,
                                                          aci, y2);
  k_gate<<<TOKENS, 256, 0, stream>>>(zxb, xin, Dv, norm_w, y2);
  k_gemm<<<dim3(DMODEL / 64, TOKENS / 64), 256, 0, stream>>>(y2, W_out, out,
                                                             1024, DMODEL);
}